// RemezArcFaceLoss_21844203667631
// MI455X (gfx1250) — compile-verified
//
#include <hip/hip_runtime.h>

typedef float v2f __attribute__((ext_vector_type(2)));
typedef float v4f __attribute__((ext_vector_type(4)));
typedef float v8f __attribute__((ext_vector_type(8)));

#define NUM_CLASSES 50000
#define BATCH       2048
#define SCALE_F     30.0f
#define EPS_F       1e-7f
// exp(SCALE*x) == exp2(SCALE*log2(e)*x) -> single v_exp_f32
#define K_EXP2      43.28085122666891f   // 30 * log2(e)

// ---------------------------------------------------------------------------
// Kernel 1: one workgroup (256 threads = 8 wave32) per row of [2048, 50000].
// Streams the row with non-temporal b128 loads (409.6 MB total, used once ->
// NT keeps it out of L2), exponentiates with raw v_exp_f32, and accumulates
// via V_WMMA_F32_16X16X4_F32 with an all-ones A matrix:
//   D[m][n] = C[m][n] + sum_k B[k][n]
// Summing all 256 accumulator elements at the end yields 16x the sum of every
// value fed through B, independent of B's lane->(k,n) mapping.
// Main loop is guard-free (48 full iterations of 1024 floats); the 212-float4
// tail is peeled and handled with lane masks (WMMA itself stays wave-uniform,
// EXEC all-ones). Two accumulators break the WMMA->WMMA RAW chain.
// Writes others[row] = rowsum - exp(SCALE*logits[row, labels[row]]).
// ---------------------------------------------------------------------------
__global__ __launch_bounds__(256) void
arcface_rowsum_kernel(const float* __restrict__ logits,
                      const int*   __restrict__ labels,
                      float*       __restrict__ others)
{
    const int    row  = blockIdx.x;
    const int    tid  = threadIdx.x;
    const size_t base = (size_t)row * (size_t)NUM_CLASSES;

    const v4f* __restrict__ rowp = (const v4f*)(logits + base);
    const int n4   = NUM_CLASSES / 4;   // 12500 float4 per row
    const int FULL = n4 / 256;          // 48 guard-free iterations

    const v2f a_ones = {1.0f, 1.0f};    // A = ones(16x4)
    v8f c0 = {0.f, 0.f, 0.f, 0.f, 0.f, 0.f, 0.f, 0.f};
    v8f c1 = {0.f, 0.f, 0.f, 0.f, 0.f, 0.f, 0.f, 0.f};

    #pragma unroll 2
    for (int it = 0; it < FULL; ++it) {
        const v4f v = __builtin_nontemporal_load(rowp + it * 256 + tid);
        v2f b0, b1;
        b0.x = __builtin_amdgcn_exp2f(K_EXP2 * v.x);
        b0.y = __builtin_amdgcn_exp2f(K_EXP2 * v.y);
        b1.x = __builtin_amdgcn_exp2f(K_EXP2 * v.z);
        b1.y = __builtin_amdgcn_exp2f(K_EXP2 * v.w);
        c0 = __builtin_amdgcn_wmma_f32_16x16x4_f32(false, a_ones, false, b0,
                                                   (short)0, c0, false, false);
        c1 = __builtin_amdgcn_wmma_f32_16x16x4_f32(false, a_ones, false, b1,
                                                   (short)0, c1, false, false);
    }

    // Peeled tail: j in [12288, 12500), 212 of 256 lanes active for the load;
    // WMMA executes wave-uniformly with zeros in masked lanes.
    {
        const int  j = FULL * 256 + tid;
        const bool m = (j < n4);
        v4f v = {0.f, 0.f, 0.f, 0.f};
        if (m) v = __builtin_nontemporal_load(rowp + j);
        v2f b0, b1;
        b0.x = m ? __builtin_amdgcn_exp2f(K_EXP2 * v.x) : 0.0f;
        b0.y = m ? __builtin_amdgcn_exp2f(K_EXP2 * v.y) : 0.0f;
        b1.x = m ? __builtin_amdgcn_exp2f(K_EXP2 * v.z) : 0.0f;
        b1.y = m ? __builtin_amdgcn_exp2f(K_EXP2 * v.w) : 0.0f;
        c0 = __builtin_amdgcn_wmma_f32_16x16x4_f32(false, a_ones, false, b0,
                                                   (short)0, c0, false, false);
        c1 = __builtin_amdgcn_wmma_f32_16x16x4_f32(false, a_ones, false, b1,
                                                   (short)0, c1, false, false);
    }

    // per-lane share of the 16x-redundant accumulators
    float s = (c0[0] + c0[1]) + (c0[2] + c0[3]) + (c0[4] + c0[5]) + (c0[6] + c0[7])
            + (c1[0] + c1[1]) + (c1[2] + c1[3]) + (c1[4] + c1[5]) + (c1[6] + c1[7]);

    // wave32 butterfly reduction
    #pragma unroll
    for (int off = 16; off >= 1; off >>= 1)
        s += __shfl_xor(s, off, 32);
    s *= (1.0f / 16.0f);                // undo 16x row redundancy of D

    __shared__ float wsum[8];
    const int wave = tid >> 5;
    const int lane = tid & 31;
    if (lane == 0) wsum[wave] = s;
    __syncthreads();

    if (tid == 0) {
        float tot = 0.0f;
        #pragma unroll
        for (int w = 0; w < 8; ++w) tot += wsum[w];
        const int   lab  = labels[row];
        const float texp = __builtin_amdgcn_exp2f(K_EXP2 * logits[base + (size_t)lab]);
        others[row] = tot - texp;       // full rowsum minus target column
    }
}

// ---------------------------------------------------------------------------
// Kernel 2: 2048-element epilogue. Gather logits[lab,lab], clamp, Horner
// degree-6 poly, L = num - log(exp(num) + others), out = -mean(L).
// ---------------------------------------------------------------------------
__global__ __launch_bounds__(256) void
arcface_finalize_kernel(const float* __restrict__ logits,
                        const int*   __restrict__ labels,
                        const float* __restrict__ coeffs,
                        const float* __restrict__ others,
                        float*       __restrict__ out)
{
    __shared__ float red[256];
    const int tid = threadIdx.x;

    // 7 descending-order coefficients (uniform -> scalar loads)
    const float k0 = coeffs[0], k1 = coeffs[1], k2 = coeffs[2], k3 = coeffs[3];
    const float k4 = coeffs[4], k5 = coeffs[5], k6 = coeffs[6];

    float acc = 0.0f;
    for (int i = tid; i < BATCH; i += 256) {
        const int   lab = labels[i];
        float cos_t = logits[(size_t)lab * (size_t)NUM_CLASSES + (size_t)lab];
        cos_t = fminf(fmaxf(cos_t, -1.0f + EPS_F), 1.0f - EPS_F);
        // jnp.clip(x, -1, 1) afterwards is a no-op given the tighter clip above
        float p = k0;
        p = __builtin_fmaf(p, cos_t, k1);
        p = __builtin_fmaf(p, cos_t, k2);
        p = __builtin_fmaf(p, cos_t, k3);
        p = __builtin_fmaf(p, cos_t, k4);
        p = __builtin_fmaf(p, cos_t, k5);
        p = __builtin_fmaf(p, cos_t, k6);
        const float num   = SCALE_F * p;
        const float denom = expf(num) + others[i];
        acc += num - logf(denom);
    }

    red[tid] = acc;
    __syncthreads();
    #pragma unroll
    for (int off = 128; off >= 1; off >>= 1) {
        if (tid < off) red[tid] += red[tid + off];
        __syncthreads();
    }
    if (tid == 0) out[0] = -red[0] * (1.0f / (float)BATCH);
}

// ---------------------------------------------------------------------------
extern "C" void kernel_launch(void* const* d_in, const int* in_sizes, int n_in,
                              void* d_out, int out_size, void* d_ws, size_t ws_size,
                              hipStream_t stream)
{
    const float* logits = (const float*)d_in[0];   // [2048*50000] f32
    const int*   labels = (const int*)d_in[1];     // [2048] i32
    const float* coeffs = (const float*)d_in[2];   // [7] f32, descending
    float*       out    = (float*)d_out;           // [1] f32
    float*       others = (float*)d_ws;            // [2048] f32 scratch

    arcface_rowsum_kernel<<<BATCH, 256, 0, stream>>>(logits, labels, others);
    arcface_finalize_kernel<<<1, 256, 0, stream>>>(logits, labels, coeffs, others, out);
}